// SAGE_43593918054565
// MI455X (gfx1250) — compile-verified
//
#include <hip/hip_runtime.h>
#include <hip/hip_bf16.h>
#include <math.h>

// ---------------------------------------------------------------------------
// SAGEConv(mean, normalize=True) for MI455X / gfx1250.
//   agg/cnt scatter : HW float atomics (memory-bound phase, ~1 GB traffic)
//   fused GEMM      : V_WMMA_F32_16X16X4_F32, K=256 = [mean | x] @ [Wl^T|Wr^T]
//   epilogue        : L2-normalize per row (shfl_xor reduce) + ReLU
// ---------------------------------------------------------------------------

typedef __attribute__((ext_vector_type(2))) float v2f;
typedef __attribute__((ext_vector_type(8))) float v8f;

#define D 128
#define WSTRIDE 132                 // padded LDS row stride (bank-conflict free)
#define WR_OFF (128 * WSTRIDE)      // offset of W_r region in LDS

__global__ void sage_zero(float* __restrict__ p, int n) {
  int i = blockIdx.x * blockDim.x + threadIdx.x;
  int stride = gridDim.x * blockDim.x;
  for (; i < n; i += stride) p[i] = 0.0f;
}

// One wave (32 lanes) per edge; each lane owns 4 contiguous floats of the row.
__global__ void sage_scatter(const float* __restrict__ x, const int* __restrict__ ei,
                             float* __restrict__ agg, float* __restrict__ cnt, int nE) {
  int gid = blockIdx.x * blockDim.x + threadIdx.x;
  int e = gid >> 5;
  int lane = gid & 31;
  if (e >= nE) return;
  int src = ei[e];
  int dst = ei[nE + e];
  const float4 v = *(const float4*)(x + (size_t)src * D + lane * 4);
  float* a = agg + (size_t)dst * D + lane * 4;
  unsafeAtomicAdd(a + 0, v.x);
  unsafeAtomicAdd(a + 1, v.y);
  unsafeAtomicAdd(a + 2, v.z);
  unsafeAtomicAdd(a + 3, v.w);
  if (lane == 0) unsafeAtomicAdd(cnt + dst, 1.0f);
}

// 256 threads = 8 waves; each wave computes a 16-node x 128-feature output tile.
__launch_bounds__(256)
__global__ void sage_gemm(const float* __restrict__ x, const float* __restrict__ Wl,
                          const float* __restrict__ bl, const float* __restrict__ Wr,
                          const float* __restrict__ agg, const float* __restrict__ cnt,
                          float* __restrict__ out, int nNodes, int nTiles) {
  extern __shared__ float ldsw[];  // [2][128 rows][132 floats]; 132-stride => 0 bank conflicts

  // Cooperative stage of both weight matrices into LDS (coalesced reads).
  for (int idx = threadIdx.x; idx < D * D; idx += blockDim.x) {
    int n = idx >> 7, k = idx & 127;
    ldsw[n * WSTRIDE + k]          = Wl[idx];
    ldsw[WR_OFF + n * WSTRIDE + k] = Wr[idx];
  }
  __syncthreads();

  int wave = blockIdx.x * (blockDim.x >> 5) + (threadIdx.x >> 5);
  if (wave >= nTiles) return;
  int lane  = threadIdx.x & 31;
  int nl    = lane & 15;            // column-class / A-row within tile
  int khalf = (lane >> 4) << 1;     // lanes 16-31 carry K+2,K+3 of each A/B step
  int base  = wave * 16;

  int arowi = base + nl;            // A-matrix row this lane feeds
  if (arowi > nNodes - 1) arowi = nNodes - 1;   // clamp (reads only; writes guarded)
  const float* arow = agg + (size_t)arowi * D;
  const float* xrow = x   + (size_t)arowi * D;
  float invc = 1.0f / fmaxf(cnt[arowi], 1.0f);

  // C init: bias b_l broadcast per output column (N = t*16 + nl for this lane).
  v8f acc[8];
#pragma unroll
  for (int t = 0; t < 8; ++t) {
    float bv = bl[t * 16 + nl];
#pragma unroll
    for (int j = 0; j < 8; ++j) acc[t][j] = bv;
  }

  // K = 0..127 : mean-half, B from W_l
  for (int ks = 0; ks < 32; ++ks) {
    int k0 = ks * 4 + khalf;
    v2f a;
    a.x = arow[k0] * invc;
    a.y = arow[k0 + 1] * invc;
#pragma unroll
    for (int t = 0; t < 8; ++t) {
      const float* bp = ldsw + (t * 16 + nl) * WSTRIDE + k0;
      v2f b; b.x = bp[0]; b.y = bp[1];
      acc[t] = __builtin_amdgcn_wmma_f32_16x16x4_f32(false, a, false, b,
                                                     (short)0, acc[t], false, false);
    }
  }
  // K = 128..255 : x-half, B from W_r
  for (int ks = 0; ks < 32; ++ks) {
    int k0 = ks * 4 + khalf;
    v2f a;
    a.x = xrow[k0];
    a.y = xrow[k0 + 1];
#pragma unroll
    for (int t = 0; t < 8; ++t) {
      const float* bp = ldsw + WR_OFF + (t * 16 + nl) * WSTRIDE + k0;
      v2f b; b.x = bp[0]; b.y = bp[1];
      acc[t] = __builtin_amdgcn_wmma_f32_16x16x4_f32(false, a, false, b,
                                                     (short)0, acc[t], false, false);
    }
  }

  // L2-normalize each output row. C layout: VGPR j holds row j (lanes 0-15) /
  // row j+8 (lanes 16-31), column N = t*16 + (lane&15). shfl_xor masks 1..8
  // reduce within each 16-lane half, giving each half its own row's sum.
  float inv[8];
#pragma unroll
  for (int j = 0; j < 8; ++j) {
    float s = 0.0f;
#pragma unroll
    for (int t = 0; t < 8; ++t) { float v = acc[t][j]; s += v * v; }
    s += __shfl_xor(s, 1);
    s += __shfl_xor(s, 2);
    s += __shfl_xor(s, 4);
    s += __shfl_xor(s, 8);
    inv[j] = 1.0f / fmaxf(sqrtf(s), 1e-12f);
  }

  int rbase = base + ((lane >> 4) << 3);   // rows j (lanes 0-15) / j+8 (lanes 16-31)
#pragma unroll
  for (int j = 0; j < 8; ++j) {
    int r = rbase + j;
    if (r < nNodes) {
      float* orow = out + (size_t)r * D + nl;
#pragma unroll
      for (int t = 0; t < 8; ++t)
        orow[t * 16] = fmaxf(acc[t][j] * inv[j], 0.0f);
    }
  }
}

extern "C" void kernel_launch(void* const* d_in, const int* in_sizes, int n_in,
                              void* d_out, int out_size, void* d_ws, size_t ws_size,
                              hipStream_t stream) {
  const float* x  = (const float*)d_in[0];
  const int*   ei = (const int*)d_in[1];   // [2, E] int32 (src row, dst row)
  const float* Wl = (const float*)d_in[2];
  const float* bl = (const float*)d_in[3];
  const float* Wr = (const float*)d_in[4];
  float* out = (float*)d_out;

  int nNodes = in_sizes[0] / D;            // 50000
  int nE     = in_sizes[1] / 2;            // 800000

  // Workspace: agg [nNodes*128] f32, cnt [nNodes] f32 (~25.8 MB)
  float* agg = (float*)d_ws;
  float* cnt = agg + (size_t)nNodes * D;

  int zn = nNodes * D + nNodes;
  sage_zero<<<2048, 256, 0, stream>>>(agg, zn);

  int scatterBlocks = (nE * 32 + 255) / 256;
  sage_scatter<<<scatterBlocks, 256, 0, stream>>>(x, ei, agg, cnt, nE);

  int nTiles = (nNodes + 15) / 16;
  int gemmBlocks = (nTiles + 7) / 8;
  size_t ldsBytes = (size_t)(2 * 128 * WSTRIDE) * sizeof(float);  // 132 KB
  sage_gemm<<<gemmBlocks, 256, ldsBytes, stream>>>(x, Wl, bl, Wr, agg, cnt, out,
                                                   nNodes, nTiles);
}